// MultiQueryAttention_86380382257376
// MI455X (gfx1250) — compile-verified
//
#include <hip/hip_runtime.h>

#define D_MODEL 512
#define SEQ     2048
#define BATCH   2

typedef __attribute__((ext_vector_type(2))) float v2f;
typedef __attribute__((ext_vector_type(8))) float v8f;

// d_ws layout (float offsets)
#define WS_XPART 0        // [2][16][512]  partial column sums of x
#define WS_XSUMP 16384    // [16][512]     zero-padded A for gemm1 (rows 0..1 valid)
#define WS_VSUMP 24576    // [16][512]     zero-padded A for gemm2 (rows 0..1 valid)
#define WS_RPART 32768    // [8][2][512]   K-chunk partials of final row
#define WS_ROW   40960    // [2][512]      final per-batch output row

// ---- Stage 1: partial sums of x over seq (64 blocks x 256 thr, coalesced) ----
__global__ void k_xpart(const float* __restrict__ x, float* __restrict__ ws) {
    const int b   = blockIdx.x >> 5;          // 0..1
    const int rem = blockIdx.x & 31;
    const int sc  = rem >> 1;                 // 0..15  (128-row chunk)
    const int col = (rem & 1) * 256 + threadIdx.x;
    const float* p = x + ((size_t)(b * SEQ + sc * 128)) * D_MODEL + col;
    float acc = 0.f;
#pragma unroll 4
    for (int s = 0; s < 128; ++s) acc += p[(size_t)s * D_MODEL];
    ws[WS_XPART + (b * 16 + sc) * D_MODEL + col] = acc;
}

// ---- Stage 2: reduce partials -> zero-padded 16x512 A matrix ----
__global__ void k_xsum_pad(float* __restrict__ ws) {
    const int g = blockIdx.x * blockDim.x + threadIdx.x;  // 0..8191
    const int m = g >> 9, col = g & 511;
    float v = 0.f;
    if (m < BATCH) {
#pragma unroll
        for (int sc = 0; sc < 16; ++sc) v += ws[WS_XPART + (m * 16 + sc) * D_MODEL + col];
    }
    ws[WS_XSUMP + g] = v;
}

// ---- Stage 3: vsumPad[16x512] = xsumPad @ Wv  (+ SEQ*bv on rows 0..1) ----
// V_WMMA_F32_16X16X4_F32, one wave per 16-col tile, K loop over 512 in steps of 4.
__global__ void k_gemm1(const float* __restrict__ Wv, const float* __restrict__ bv,
                        float* __restrict__ ws) {
    const int lane  = threadIdx.x & 31;
    const int wave  = threadIdx.x >> 5;
    const int ntile = blockIdx.x * 8 + wave;      // 0..31 -> 512 cols
    const int col   = ntile * 16 + (lane & 15);
    const int half  = lane >> 4;                  // lane group (K pairing per ISA layout)
    const int m     = lane & 15;                  // A-matrix row for this lane
    const float* A  = ws + WS_XSUMP;
    v8f acc = {};
    for (int kb = 0; kb < D_MODEL; kb += 4) {
        const int k0 = kb + half * 2;
        // A 16x4: lane holds a[m][k0], a[m][k0+1] in VGPR0/1 (ISA 7.12.2)
        v2f a = { A[m * D_MODEL + k0], A[m * D_MODEL + k0 + 1] };
        // B 4x16: rows striped across lanes within a VGPR, mirrored K pairing
        v2f bm = { Wv[(size_t)k0 * D_MODEL + col],
                   Wv[(size_t)(k0 + 1) * D_MODEL + col] };
        acc = __builtin_amdgcn_wmma_f32_16x16x4_f32(false, a, false, bm,
                                                    (short)0, acc, false, false);
    }
    // C/D layout: VGPR r, lanes 0-15 -> M=r, lanes 16-31 -> M=8+r
    float* V = ws + WS_VSUMP;
#pragma unroll
    for (int r = 0; r < 8; ++r) {
        const int row = half * 8 + r;
        float val = acc[r];
        if (row < BATCH) val += (float)SEQ * bv[col];   // bias only on real rows
        V[row * D_MODEL + col] = val;                   // rows >=2 stay exactly 0
    }
}

// ---- Stage 4: rpart[kc] = tile_8(vsum) @ Wo (K-chunk kc of 8) ----
// tile_8(vsum)[k] = vsum[k mod 512], and each 512-wide chunk aligns with the tiling,
// so every chunk reuses the same 16x512 A matrix.
__global__ void k_gemm2(const float* __restrict__ Wo, float* __restrict__ ws) {
    const int lane  = threadIdx.x & 31;
    const int wave  = threadIdx.x >> 5;
    const int nblk  = blockIdx.x & 3;
    const int kc    = blockIdx.x >> 2;            // 0..7 K-chunk
    const int ntile = nblk * 8 + wave;            // 0..31 -> 512 cols
    const int col   = ntile * 16 + (lane & 15);
    const int half  = lane >> 4;
    const int m     = lane & 15;
    const float* A  = ws + WS_VSUMP;
    const float* B  = Wo + (size_t)kc * 512 * D_MODEL;
    v8f acc = {};
    for (int kb = 0; kb < 512; kb += 4) {
        const int k0 = kb + half * 2;
        v2f a = { A[m * D_MODEL + k0], A[m * D_MODEL + k0 + 1] };
        v2f bm = { B[(size_t)k0 * D_MODEL + col],
                   B[(size_t)(k0 + 1) * D_MODEL + col] };
        acc = __builtin_amdgcn_wmma_f32_16x16x4_f32(false, a, false, bm,
                                                    (short)0, acc, false, false);
    }
    if (half == 0) {   // M=0 -> acc[0], M=1 -> acc[1] (lanes 0-15)
        ws[WS_RPART + (kc * 2 + 0) * D_MODEL + col] = acc[0];
        ws[WS_RPART + (kc * 2 + 1) * D_MODEL + col] = acc[1];
    }
}

// ---- Stage 5: reduce K-chunk partials + bo -> final 2x512 row ----
__global__ void k_row(const float* __restrict__ bo, float* __restrict__ ws) {
    const int g = blockIdx.x * blockDim.x + threadIdx.x;  // 0..1023
    const int b = g >> 9, col = g & 511;
    float v = bo[col];
#pragma unroll
    for (int kc = 0; kc < 8; ++kc) v += ws[WS_RPART + (kc * 2 + b) * D_MODEL + col];
    ws[WS_ROW + g] = v;
}

// ---- Stage 6: broadcast row over seq (the dominant 8 MB write, float4) ----
__global__ void k_bcast(const float* __restrict__ ws, float* __restrict__ out) {
    const int g    = blockIdx.x * blockDim.x + threadIdx.x;  // float4 index
    const int col4 = g & 127;          // 512/4 float4 per row
    const int row  = g >> 7;           // 0..4095 (b*2048 + s)
    const int b    = row >> 11;
    const float4* src = (const float4*)(ws + WS_ROW + b * D_MODEL);
    ((float4*)out)[g] = src[col4];
}

extern "C" void kernel_launch(void* const* d_in, const int* in_sizes, int n_in,
                              void* d_out, int out_size, void* d_ws, size_t ws_size,
                              hipStream_t stream) {
    (void)in_sizes; (void)n_in; (void)out_size; (void)ws_size;
    const float* x  = (const float*)d_in[0];
    // d_in[1..4] = Wq, bq, Wk, bk: provably unused — the 'bqnk,bvd->bqnd' einsum
    // factorizes so softmax rows contribute exactly their sum (== 1).
    const float* Wv = (const float*)d_in[5];
    const float* bv = (const float*)d_in[6];
    const float* Wo = (const float*)d_in[7];
    const float* bo = (const float*)d_in[8];
    float* ws  = (float*)d_ws;
    float* out = (float*)d_out;

    k_xpart   <<<  64, 256, 0, stream>>>(x, ws);
    k_xsum_pad<<<  32, 256, 0, stream>>>(ws);
    k_gemm1   <<<   4, 256, 0, stream>>>(Wv, bv, ws);
    k_gemm2   <<<  32, 256, 0, stream>>>(Wo, ws);
    k_row     <<<   4, 256, 0, stream>>>(bo, ws);
    k_bcast   <<<2048, 256, 0, stream>>>(ws, out);
}